// Camera_53231824666915
// MI455X (gfx1250) — compile-verified
//
#include <hip/hip_runtime.h>
#include <math.h>

typedef __attribute__((ext_vector_type(2))) float v2f;
typedef __attribute__((ext_vector_type(8))) float v8f;

// Fast reciprocal: v_rcp_f32 + one Newton-Raphson step (~1 ulp).
// Avoids the ~10-instruction precise-division expansion (div_scale/div_fmas/
// div_fixup) per divide; this kernel needs 2 reciprocals per point.
__device__ __forceinline__ float fast_rcp(float a)
{
    float r = __builtin_amdgcn_rcpf(a);
    r = __builtin_fmaf(__builtin_fmaf(-a, r, 1.0f), r, r);
    return r;
}

// Per-point epilogue: perspective divide, frustum mask, pos2d, cov2d, stores.
__device__ __forceinline__ void cam_epilogue(
    float c0, float c1, float c2, float c3,
    float x, float y, float z,
    const float* __restrict__ cov,
    float R00, float R01, float R02,
    float R10, float R11, float R12,
    float R20, float R21, float R22,
    float* __restrict__ out, long long idx, long long n)
{
    const float w    = c3 + 1e-6f;
    const float invw = fast_rcp(w);
    const float ndx = c0 * invw;
    const float ndy = c1 * invw;
    const float ndz = c2 * invw;

    const bool m = (ndz >= 0.2f) && (ndx >= -1.3f) && (ndx <= 1.3f)
                                 && (ndy >= -1.3f) && (ndy <= 1.3f);

    const float px = 0.5f * (ndx + 1.0f) * 1920.0f;
    const float py = (1.0f - 0.5f * (ndy + 1.0f)) * 1080.0f;

    // M = J * R^T rows:  u_k = iz*(R[k][0] - x*R[k][2]),  v_k = iz*(R[k][1] - y*R[k][2])
    const float iz = fast_rcp(z);
    const float u0 = iz * (R00 - x * R02);
    const float u1 = iz * (R10 - x * R12);
    const float u2 = iz * (R20 - x * R22);
    const float v0 = iz * (R01 - y * R02);
    const float v1 = iz * (R11 - y * R12);
    const float v2 = iz * (R21 - y * R22);

    const size_t cb = (size_t)idx * 9;
    const float s00 = __builtin_nontemporal_load(&cov[cb + 0]);
    const float s01 = __builtin_nontemporal_load(&cov[cb + 1]);
    const float s02 = __builtin_nontemporal_load(&cov[cb + 2]);
    const float s10 = __builtin_nontemporal_load(&cov[cb + 3]);
    const float s11 = __builtin_nontemporal_load(&cov[cb + 4]);
    const float s12 = __builtin_nontemporal_load(&cov[cb + 5]);
    const float s20 = __builtin_nontemporal_load(&cov[cb + 6]);
    const float s21 = __builtin_nontemporal_load(&cov[cb + 7]);
    const float s22 = __builtin_nontemporal_load(&cov[cb + 8]);

    const float cu0 = s00 * u0 + s01 * u1 + s02 * u2;
    const float cu1 = s10 * u0 + s11 * u1 + s12 * u2;
    const float cu2 = s20 * u0 + s21 * u1 + s22 * u2;
    const float cv0 = s00 * v0 + s01 * v1 + s02 * v2;
    const float cv1 = s10 * v0 + s11 * v1 + s12 * v2;
    const float cv2 = s20 * v0 + s21 * v1 + s22 * v2;

    const float q00 = u0 * cu0 + u1 * cu1 + u2 * cu2;
    const float q01 = u0 * cv0 + u1 * cv1 + u2 * cv2;
    const float q10 = v0 * cu0 + v1 * cu1 + v2 * cu2;
    const float q11 = v0 * cv0 + v1 * cv1 + v2 * cv2;

    const size_t pb = (size_t)idx * 3;
    __builtin_nontemporal_store(m ? px  : 0.0f, &out[pb + 0]);
    __builtin_nontemporal_store(m ? py  : 0.0f, &out[pb + 1]);
    __builtin_nontemporal_store(m ? ndz : 0.0f, &out[pb + 2]);

    const size_t qb = (size_t)(3 * n) + (size_t)idx * 4;
    __builtin_nontemporal_store(m ? q00 : 0.0f, &out[qb + 0]);
    __builtin_nontemporal_store(m ? q01 : 0.0f, &out[qb + 1]);
    __builtin_nontemporal_store(m ? q10 : 0.0f, &out[qb + 2]);
    __builtin_nontemporal_store(m ? q11 : 0.0f, &out[qb + 3]);

    __builtin_nontemporal_store(m ? 1.0f : 0.0f, &out[(size_t)(7 * n) + (size_t)idx]);
}

__global__ void __launch_bounds__(256)
cam_proj_kernel(const float* __restrict__ pos, const float* __restrict__ cov,
                const float* __restrict__ Wc, float* __restrict__ out,
                long long n, float p00, float p11, float p22, float p23)
{
    const int  lane = threadIdx.x & 31;
    const int  r    = lane & 15;
    const bool hi   = lane >= 16;

    // Wave index as a *compiler-known scalar* so loop control compiles to
    // s_cmp/s_cbranch (no exec-mask dance) and EXEC is provably all-1s at
    // the WMMAs.
    const int waveInBlock = __builtin_amdgcn_readfirstlane((int)(threadIdx.x >> 5));
    const int wavesPerBlk = (int)(blockDim.x >> 5);

    // ---- Hoisted, loop-invariant setup (amortized over the grid-stride loop) ----
    // Uniform world2model (rows of R, translation t).
    const float R00 = Wc[0], R01 = Wc[1], R02 = Wc[2],  t0 = Wc[3];
    const float R10 = Wc[4], R11 = Wc[5], R12 = Wc[6],  t1 = Wc[7];
    const float R20 = Wc[8], R21 = Wc[9], R22 = Wc[10], t2 = Wc[11];

    // T = P * W2C (P is sparse: only p00,p11,p22,p23 and P[3][2]=1 nonzero).
    const float T00 = p00 * R00, T01 = p00 * R01, T02 = p00 * R02, T03 = p00 * t0;
    const float T10 = p11 * R10, T11 = p11 * R11, T12 = p11 * R12, T13 = p11 * t1;
    const float T20 = p22 * R20, T21 = p22 * R21, T22 = p22 * R22, T23 = p22 * t2 + p23;
    const float T30 = R20,       T31 = R21,       T32 = R22,       T33 = t2;

    // WMMA A operand (16x4 f32, loop-invariant): row m = clip component
    // (rows 4..15 zero).  K layout: lanes 0-15 -> K=0(a0),K=1(a1);
    // lanes 16-31 -> K=2(a0),K=3(a1).
    const float a0 = (r == 0) ? (hi ? T02 : T00)
                   : (r == 1) ? (hi ? T12 : T10)
                   : (r == 2) ? (hi ? T22 : T20)
                   : (r == 3) ? (hi ? T32 : T30) : 0.0f;
    const float a1 = (r == 0) ? (hi ? T03 : T01)
                   : (r == 1) ? (hi ? T13 : T11)
                   : (r == 2) ? (hi ? T23 : T21)
                   : (r == 3) ? (hi ? T33 : T31) : 0.0f;
    v2f A; A.x = a0; A.y = a1;
    v8f acc = {};

    // ---- Grid-stride loop over full 32-point tiles ----
    const long long wavesTotal = (long long)gridDim.x * (long long)wavesPerBlk;
    const long long waveId =
        (long long)blockIdx.x * (long long)wavesPerBlk + (long long)waveInBlock;
    const long long tileStride = wavesTotal * 32LL;

    long long base = waveId * 32LL;
    for (; base + 32 <= n; base += tileStride) {
        // EXEC all-1s here (scalar loop bound) as WMMA requires.
        const long long idx = base + lane;
        const size_t pb = (size_t)idx * 3;
        const float x = __builtin_nontemporal_load(&pos[pb + 0]);
        const float y = __builtin_nontemporal_load(&pos[pb + 1]);
        const float z = __builtin_nontemporal_load(&pos[pb + 2]);

        // B operand #1 (4x16 f32): column n = point base+n.
        // lanes 0-15: K=0 -> x, K=1 -> y ; lanes 16-31: K=2 -> z, K=3 -> 1.
        const float zlo = __shfl(z, r, 32);       // z of point (lane&15)
        v2f B1; B1.x = hi ? zlo : x; B1.y = hi ? 1.0f : y;
        v8f D1 = __builtin_amdgcn_wmma_f32_16x16x4_f32(
            false, A, false, B1, (short)0, acc, false, false);

        // B operand #2: points base+16 .. base+31.
        const float xhi = __shfl(x, r + 16, 32);  // x of point (lane&15)+16
        const float yhi = __shfl(y, r + 16, 32);
        v2f B2; B2.x = hi ? z : xhi; B2.y = hi ? 1.0f : yhi;
        v8f D2 = __builtin_amdgcn_wmma_f32_16x16x4_f32(
            false, A, false, B2, (short)0, acc, false, false);

        // D[m][n]: VGPR m (m<8), lane n (n<16) = clip component m of point n.
        // Lanes 16-31 pull point (16 + lane&15) from the second WMMA result.
        const float s0 = __shfl(D2[0], r, 32);
        const float s1 = __shfl(D2[1], r, 32);
        const float s2 = __shfl(D2[2], r, 32);
        const float s3 = __shfl(D2[3], r, 32);
        const float c0 = hi ? s0 : D1[0];
        const float c1 = hi ? s1 : D1[1];
        const float c2 = hi ? s2 : D1[2];
        const float c3 = hi ? s3 : D1[3];

        cam_epilogue(c0, c1, c2, c3, x, y, z, cov,
                     R00, R01, R02, R10, R11, R12, R20, R21, R22,
                     out, idx, n);
    }

    // ---- Single partial tail tile (at most one per grid): plain per-lane path ----
    if (base < n) {
        const long long idx = base + lane;
        if (idx < n) {
            const size_t pb = (size_t)idx * 3;
            const float x = pos[pb + 0];
            const float y = pos[pb + 1];
            const float z = pos[pb + 2];
            const float c0 = T00 * x + T01 * y + T02 * z + T03;
            const float c1 = T10 * x + T11 * y + T12 * z + T13;
            const float c2 = T20 * x + T21 * y + T22 * z + T23;
            const float c3 = T30 * x + T31 * y + T32 * z + T33;
            cam_epilogue(c0, c1, c2, c3, x, y, z, cov,
                         R00, R01, R02, R10, R11, R12, R20, R21, R22,
                         out, idx, n);
        }
    }
}

extern "C" void kernel_launch(void* const* d_in, const int* in_sizes, int n_in,
                              void* d_out, int out_size, void* d_ws, size_t ws_size,
                              hipStream_t stream)
{
    const float* pos = (const float*)d_in[0];   // (N,3) f32
    const float* cov = (const float*)d_in[1];   // (N,3,3) f32
    const float* w2c = (const float*)d_in[2];   // (4,4) f32
    float* out = (float*)d_out;                 // pos2d(3N) | cov2d(4N) | mask(N)

    const long long n = (long long)(in_sizes[0] / 3);

    // Projection-matrix constants (host-side, exact):
    const double ZNEAR = 0.1, ZFAR = 100.0;
    const float p00 = (float)(1.0 / tan(0.5 * 1.2));           // 2*zn/(r-l), l=-r
    const float p11 = (float)(1.0 / tan(0.5 * 0.9));
    const float p22 = (float)(ZFAR / (ZFAR - ZNEAR));
    const float p23 = (float)(-ZFAR * ZNEAR / (ZFAR - ZNEAR));

    const int  threads = 256;                                  // 8 waves/block
    const long long waves    = (n + 31) / 32;
    long long blocksNeeded   = (waves + 7) / 8;
    long long blocks         = blocksNeeded < 1024 ? blocksNeeded : 1024;
    if (blocks < 1) blocks = 1;

    cam_proj_kernel<<<(int)blocks, threads, 0, stream>>>(
        pos, cov, w2c, out, n, p00, p11, p22, p23);
}